// ParticleQAEEncoder_62208306315381
// MI455X (gfx1250) — compile-verified
//
#include <hip/hip_runtime.h>

// ---------------------------------------------------------------------------
// Batched quantum-autoencoder forward pass for MI455X (gfx1250, wave32).
//   met : closed form.   ele/mu : 16 amps in registers, 1 thread/event.
//   jet : 1024 amps in wave-private LDS, 1 wave/event, gates fused to
//         3-qubit 16x16 real matrices executed with V_WMMA_F32_16X16X4_F32
//         chains (K=16, 100% utilization).  CNOT blocks collapse to
//         parity-controlled bitmask swaps (pure LDS permutation).
// ---------------------------------------------------------------------------

typedef float v2f __attribute__((ext_vector_type(2)));
typedef float v8f __attribute__((ext_vector_type(8)));

#if defined(__has_builtin)
#  if __has_builtin(__builtin_amdgcn_wmma_f32_16x16x4_f32)
#    define HAVE_WMMA_F32F32 1
#  endif
#endif

struct cplx { float re, im; };
__device__ __forceinline__ cplx cmul(cplx a, cplx b){
  return { fmaf(a.re, b.re, -(a.im * b.im)), fmaf(a.re, b.im, a.im * b.re) };
}
__device__ __forceinline__ cplx cadd(cplx a, cplx b){ return { a.re + b.re, a.im + b.im }; }

struct mat2 { cplx m[2][2]; };

__device__ __forceinline__ mat2 mm2(const mat2& A, const mat2& B){
  mat2 C;
  #pragma unroll
  for (int i = 0; i < 2; ++i)
    #pragma unroll
    for (int j = 0; j < 2; ++j)
      C.m[i][j] = cadd(cmul(A.m[i][0], B.m[0][j]), cmul(A.m[i][1], B.m[1][j]));
  return C;
}

__device__ __forceinline__ mat2 rxm(float t){
  float s, c; __sincosf(t * 0.5f, &s, &c);
  mat2 M;
  M.m[0][0] = {c, 0.f};  M.m[0][1] = {0.f, -s};
  M.m[1][0] = {0.f, -s}; M.m[1][1] = {c, 0.f};
  return M;
}
__device__ __forceinline__ mat2 rym(float t){
  float s, c; __sincosf(t * 0.5f, &s, &c);
  mat2 M;
  M.m[0][0] = {c, 0.f};  M.m[0][1] = {-s, 0.f};
  M.m[1][0] = {s, 0.f};  M.m[1][1] = {c, 0.f};
  return M;
}
__device__ __forceinline__ mat2 rzm(float t){
  float s, c; __sincosf(t * 0.5f, &s, &c);
  mat2 M;
  M.m[0][0] = {c, -s};    M.m[0][1] = {0.f, 0.f};
  M.m[1][0] = {0.f, 0.f}; M.m[1][1] = {c, s};
  return M;
}
__device__ __forceinline__ mat2 identm(){
  mat2 M;
  M.m[0][0] = {1.f, 0.f}; M.m[0][1] = {0.f, 0.f};
  M.m[1][0] = {0.f, 0.f}; M.m[1][1] = {1.f, 0.f};
  return M;
}

// ------------------------- met / ele / mu (registers) ----------------------

template<int BIT>
__device__ __forceinline__ void gate4(cplx* a, const mat2 U){
  #pragma unroll
  for (int i = 0; i < 16; ++i){
    if ((i & (1 << BIT)) == 0){
      cplx a0 = a[i], a1 = a[i | (1 << BIT)];
      a[i]              = cadd(cmul(U.m[0][0], a0), cmul(U.m[0][1], a1));
      a[i | (1 << BIT)] = cadd(cmul(U.m[1][0], a0), cmul(U.m[1][1], a1));
    }
  }
}

// 4-qubit QAE block: latent wires {0,1} (bits 3,2), trash {2,3} (bits 1,0)
__device__ void qae4(const float* pt, const float* eta, const float* phi,
                     const float* w, float& o0, float& o1){
  cplx a[16];
  #pragma unroll
  for (int i = 0; i < 16; ++i) a[i] = {0.f, 0.f};
  a[0] = {1.f, 0.f};
  // encoding: U = RZ(phi) * RY(pt) * RX(eta), wire k -> bit 3-k
  gate4<3>(a, mm2(rzm(phi[0]), mm2(rym(pt[0]), rxm(eta[0]))));
  gate4<2>(a, mm2(rzm(phi[1]), mm2(rym(pt[1]), rxm(eta[1]))));
  gate4<1>(a, mm2(rzm(phi[2]), mm2(rym(pt[2]), rxm(eta[2]))));
  gate4<0>(a, mm2(rzm(phi[3]), mm2(rym(pt[3]), rxm(eta[3]))));
  // CNOT block (controls = trash bits {1,0}, targets = latent bits {3,2}):
  // flip latent mask 0xC where parity(y&3)==1  -> pairwise swap
  #pragma unroll
  for (int y = 0; y < 16; ++y){
    if ((((y & 1) ^ ((y >> 1) & 1)) == 1) && !(y & 8)){
      int z = y ^ 0xC;
      cplx t = a[y]; a[y] = a[z]; a[z] = t;
    }
  }
  gate4<3>(a, rym(w[0]));
  gate4<2>(a, rym(w[1]));
  gate4<1>(a, rym(w[2]));
  gate4<0>(a, rym(w[3]));
  // CNOT block (controls = latent bits {3,2}, targets = trash bits {1,0})
  #pragma unroll
  for (int y = 0; y < 16; ++y){
    if (((((y >> 2) & 1) ^ ((y >> 3) & 1)) == 1) && !(y & 2)){
      int z = y ^ 0x3;
      cplx t = a[y]; a[y] = a[z]; a[z] = t;
    }
  }
  o0 = 0.f; o1 = 0.f;          // <Z> on wire2 (bit1), wire3 (bit0)
  #pragma unroll
  for (int y = 0; y < 16; ++y){
    float p = a[y].re * a[y].re + a[y].im * a[y].im;
    o0 += (y & 2) ? -p : p;
    o1 += (y & 1) ? -p : p;
  }
}

__global__ void __launch_bounds__(256)
smq_kernel(const float* __restrict__ x, const float* __restrict__ met_w,
           const float* __restrict__ ele_w, const float* __restrict__ mu_w,
           float* __restrict__ out, int B){
  const int e = blockIdx.x * blockDim.x + threadIdx.x;
  if (e >= B) return;
  const float* xr = x + e * 56;
  {
    // 1-qubit block closed form: <Z> = cos(w)cos(pt) - sin(w)sin(pt)cos(phi)
    float w = met_w[0];
    out[e] = cosf(w) * cosf(xr[0]) - sinf(w) * sinf(xr[0]) * cosf(xr[1]);
  }
  float o0, o1;
  qae4(xr + 2,  xr + 6,  xr + 10, ele_w, o0, o1);
  out[B + 2 * e]     = o0;  out[B + 2 * e + 1]     = o1;
  qae4(xr + 14, xr + 18, xr + 22, mu_w,  o0, o1);
  out[3 * B + 2 * e] = o0;  out[3 * B + 2 * e + 1] = o1;
}

// ------------------------------- jet (LDS + WMMA) ---------------------------

// fallback: single-qubit gate applied per-lane on LDS (bit bb of 10-bit index)
__device__ void gate_lds(float* re, float* im, int lane, int bb, const mat2 U){
  const int m = 1 << bb;
  for (int i = 0; i < 16; ++i){
    const int p  = lane + (i << 5);                         // pair id 0..511
    const int i0 = ((p >> bb) << (bb + 1)) | (p & (m - 1));
    const int i1 = i0 | m;
    cplx a0 = { re[i0], im[i0] }, a1 = { re[i1], im[i1] };
    cplx r0 = cadd(cmul(U.m[0][0], a0), cmul(U.m[0][1], a1));
    cplx r1 = cadd(cmul(U.m[1][0], a0), cmul(U.m[1][1], a1));
    re[i0] = r0.re; im[i0] = r0.im;
    re[i1] = r1.re; im[i1] = r1.im;
  }
}

// Apply kron(V2,V1,V0) on index bits (b+2, b+1, b) of the 1024-amp state.
// Real 16x16 gate G16 = [[Gr,-Gi],[Gi,Gr]] applied as D = G16 * S via a
// K=16 chain of four V_WMMA_F32_16X16X4_F32, 16 amplitude-columns per chain.
__device__ void apply_fused3(float* re, float* im, int lane, int b,
                             const mat2 V2, const mat2 V1, const mat2 V0){
#if defined(HAVE_WMMA_F32F32)
  const int row = lane & 15;            // G16 row this lane supplies to A
  const int ch  = (lane >> 4) << 1;     // K-pair select per ISA A layout
  float a[8];
  #pragma unroll
  for (int j = 0; j < 4; ++j){
    #pragma unroll
    for (int t = 0; t < 2; ++t){
      const int col = 4 * j + ch + t;   // G16 column (0..15)
      const int rr = row & 7, cc = col & 7;
      cplx g = cmul(cmul(V2.m[(rr >> 2) & 1][(cc >> 2) & 1],
                         V1.m[(rr >> 1) & 1][(cc >> 1) & 1]),
                    V0.m[rr & 1][cc & 1]);
      float v;
      if (row < 8) v = (col < 8) ?  g.re : -g.im;
      else         v = (col < 8) ?  g.im :  g.re;
      a[2 * j + t] = v;
    }
  }
  const int lowmask = (1 << b) - 1;
  const int N = lane & 15;              // B/D column this lane owns
  for (int t = 0; t < 8; ++t){          // 8 chains x 16 columns = 128 groups
    const int col  = (t << 4) + N;
    const int base = ((col >> b) << (b + 3)) | (col & lowmask);
    v8f d = {0.f, 0.f, 0.f, 0.f, 0.f, 0.f, 0.f, 0.f};
    #pragma unroll
    for (int j = 0; j < 4; ++j){
      const float* src = (j < 2) ? re : im;            // rows 0-7 = Re, 8-15 = Im
      const int k0 = 4 * j + ch - ((j < 2) ? 0 : 8);
      v2f av; av.x = a[2 * j]; av.y = a[2 * j + 1];
      v2f bv; bv.x = src[base + (k0 << b)];
              bv.y = src[base + ((k0 + 1) << b)];
      d = __builtin_amdgcn_wmma_f32_16x16x4_f32(false, av, false, bv,
                                                (short)0, d, false, false);
    }
    // D layout: lanes 0-15 hold rows 0-7 (Re out), lanes 16-31 rows 8-15 (Im)
    float* dst = (lane < 16) ? re : im;
    #pragma unroll
    for (int v = 0; v < 8; ++v)
      dst[base + (v << b)] = d[v];
  }
#else
  gate_lds(re, im, lane, b + 2, V2);
  gate_lds(re, im, lane, b + 1, V1);
  gate_lds(re, im, lane, b,     V0);
#endif
}

// CNOT block with disjoint control/target wire sets == parity-controlled
// bitmask flip: swap amp[y] <-> amp[y^tmask] where parity(y&cmask)==1.
__device__ void perm_swap(float* re, float* im, int lane,
                          int cmask, int tmask, int pick){
  for (int i = 0; i < 32; ++i){
    const int y = lane + (i << 5);
    if ((__popc(y & cmask) & 1) && !(y & pick)){
      const int z = y ^ tmask;
      float t0 = re[y], t1 = re[z]; re[y] = t1; re[z] = t0;
      float u0 = im[y], u1 = im[z]; im[y] = u1; im[z] = u0;
    }
  }
}

__device__ __forceinline__ mat2 encU(const float* xr, int k){
  // RZ(phi) * RY(pt) * RX(eta) for jet wire k
  return mm2(rzm(xr[46 + k]), mm2(rym(xr[26 + k]), rxm(xr[36 + k])));
}

#define JWAVES 4

__global__ void __launch_bounds__(JWAVES * 32)
jet_kernel(const float* __restrict__ x, const float* __restrict__ jw,
           float* __restrict__ out, int B){
  __shared__ float sre[JWAVES * 1024];
  __shared__ float sim[JWAVES * 1024];
  const int lane = threadIdx.x & 31;
  const int wave = threadIdx.x >> 5;
  const int ev   = blockIdx.x * JWAVES + wave;
  if (ev >= B) return;
  float* re = sre + (wave << 10);       // wave-private: no barriers needed
  float* im = sim + (wave << 10);

  for (int i = 0; i < 32; ++i){
    re[lane + (i << 5)] = 0.f;
    im[lane + (i << 5)] = 0.f;
  }
  if (lane == 0) re[0] = 1.f;           // |0...0>

  const float* xr = x + ev * 56;
  // encoding layer: wire k acts on bit 9-k; fused groups of 3 bits
  apply_fused3(re, im, lane, 7, encU(xr, 0), encU(xr, 1), encU(xr, 2));
  apply_fused3(re, im, lane, 4, encU(xr, 3), encU(xr, 4), encU(xr, 5));
  apply_fused3(re, im, lane, 1, encU(xr, 6), encU(xr, 7), encU(xr, 8));
  apply_fused3(re, im, lane, 0, identm(), identm(), encU(xr, 9));

  for (int d = 0; d < 4; ++d){
    // trash(bits 3..0) controls -> latent(bits 9..4) targets
    perm_swap(re, im, lane, 0x00F, 0x3F0, 0x200);
    const float* wd = jw + d * 10;
    apply_fused3(re, im, lane, 7, rym(wd[0]), rym(wd[1]), rym(wd[2]));
    apply_fused3(re, im, lane, 4, rym(wd[3]), rym(wd[4]), rym(wd[5]));
    apply_fused3(re, im, lane, 1, rym(wd[6]), rym(wd[7]), rym(wd[8]));
    apply_fused3(re, im, lane, 0, identm(), identm(), rym(wd[9]));
    // latent controls -> trash targets
    perm_swap(re, im, lane, 0x3F0, 0x00F, 0x008);
  }

  // <Z> on trash wires 6,7,8,9  ->  bits 3,2,1,0
  float ez0 = 0.f, ez1 = 0.f, ez2 = 0.f, ez3 = 0.f;
  for (int i = 0; i < 32; ++i){
    const int y = lane + (i << 5);
    const float p = re[y] * re[y] + im[y] * im[y];
    ez0 += (y & 8) ? -p : p;
    ez1 += (y & 4) ? -p : p;
    ez2 += (y & 2) ? -p : p;
    ez3 += (y & 1) ? -p : p;
  }
  #pragma unroll
  for (int m = 16; m >= 1; m >>= 1){
    ez0 += __shfl_xor(ez0, m, 32);
    ez1 += __shfl_xor(ez1, m, 32);
    ez2 += __shfl_xor(ez2, m, 32);
    ez3 += __shfl_xor(ez3, m, 32);
  }
  if (lane == 0){
    float* o = out + 5 * B + ev * 4;
    o[0] = ez0; o[1] = ez1; o[2] = ez2; o[3] = ez3;
  }
}

// --------------------------------- launch -----------------------------------

extern "C" void kernel_launch(void* const* d_in, const int* in_sizes, int n_in,
                              void* d_out, int out_size, void* d_ws, size_t ws_size,
                              hipStream_t stream){
  (void)n_in; (void)out_size; (void)d_ws; (void)ws_size;
  const float* x     = (const float*)d_in[0];
  const float* met_w = (const float*)d_in[1];
  const float* ele_w = (const float*)d_in[2];
  const float* mu_w  = (const float*)d_in[3];
  const float* jet_w = (const float*)d_in[4];
  float* out = (float*)d_out;
  const int B = in_sizes[0] / 56;

  smq_kernel<<<(B + 255) / 256, 256, 0, stream>>>(x, met_w, ele_w, mu_w, out, B);
  jet_kernel<<<(B + JWAVES - 1) / JWAVES, JWAVES * 32, 0, stream>>>(x, jet_w, out, B);
}